// QModel_65481071403912
// MI455X (gfx1250) — compile-verified
//
#include <hip/hip_runtime.h>
#include <hip/hip_bf16.h>

// ---------------------------------------------------------------------------
// 18-qubit batched state-vector simulator for MI455X (gfx1250).
//
// Strategy:
//  * Fuse RX(0.1)*RY(0.1) per wire + intra-group CNOTs into 16x16 complex
//    unitaries over wire groups {0-3},{4-7},{8-11},{12-15}; wires {16,17}
//    get a 4x4 unitary. Cross-group CNOTs (disjoint qubit pairs) compose
//    into one involutive permutation P folded into the next pass's loads.
//  * Each 4-wire group pass is a complex GEMM (16x16) x (16 x 2^20) executed
//    with V_WMMA_F32_16X16X4_F32 (full fp32, 16 WMMAs per 16-col tile).
//  * State (128 MB) < L2 (192 MB): ping-pong passes are L2-resident.
//
// Workspace layout (floats): [0,1024) gate matrices, then 4 planes of
// 64*2^18 floats (bufA_re, bufA_im, bufB_re, bufB_im) => ~256 MB + 4 KB.
// ---------------------------------------------------------------------------

typedef float v8f __attribute__((ext_vector_type(8)));
typedef float v2f __attribute__((ext_vector_type(2)));

#define NW 18
#define DIMBITS 18
#define BATCH 64

// Cross-group CNOT ring permutation: (ctl,tgt) wires (3,4),(7,8),(11,12),
// (15,16),(17,0); wire w <-> flat bit (17-w). All ctl bits disjoint from all
// tgt bits -> single XOR-involution.
static __device__ __forceinline__ unsigned permIdx(unsigned F) {
  unsigned x = F;
  x ^= ((F >> 14) & 1u) << 13;  // ctl w3  (bit14) -> tgt w4  (bit13)
  x ^= ((F >> 10) & 1u) << 9;   // ctl w7  -> tgt w8
  x ^= ((F >> 6)  & 1u) << 5;   // ctl w11 -> tgt w12
  x ^= ((F >> 2)  & 1u) << 1;   // ctl w15 -> tgt w16
  x ^= ( F        & 1u) << 17;  // ctl w17 (bit0) -> tgt w0 (bit17)
  return x;
}

// ---------------------------------------------------------------------------
// Build fused gate matrices into ws[0..800):
//   [0,256)   G4 real      (16x16, row-major)
//   [256,512) G4 imag
//   [512,768) -G4 imag     (pre-negated for D_re accumulation)
//   [768,784) G2 real      (4x4)
//   [784,800) G2 imag
// G4 = C(q2,q3) C(q1,q2) C(q0,q1) (U (x) U (x) U (x) U),  U = RY(t)RX(t).
// Local bit3 = lowest wire of the group (matches flat-index bit order).
// ---------------------------------------------------------------------------
__global__ __launch_bounds__(256) void qsim_build(float* __restrict__ g) {
  const int t = threadIdx.x;
  const float c = cosf(0.05f), s = sinf(0.05f);
  // U = RY * RX  (RX = [[c,-is],[-is,c]], RY = [[c,-s],[s,c]])
  float ur[2][2], ui[2][2];
  {
    const float rxr[2][2] = {{c, 0.f}, {0.f, c}};
    const float rxi[2][2] = {{0.f, -s}, {-s, 0.f}};
    const float ryr[2][2] = {{c, -s}, {s, c}};
    for (int i = 0; i < 2; ++i)
      for (int j = 0; j < 2; ++j) {
        ur[i][j] = ryr[i][0] * rxr[0][j] + ryr[i][1] * rxr[1][j];
        ui[i][j] = ryr[i][0] * rxi[0][j] + ryr[i][1] * rxi[1][j];
      }
  }
  // 16x16 entry (i,j): row-permute by the 3 intra-group CNOTs.
  {
    const int i = t >> 4, j = t & 15;
    int si = i ^ ((i >> 1) & 1);          // C(q2,q3): ctl bit1 -> tgt bit0
    si ^= ((si >> 2) & 1) << 1;           // C(q1,q2): ctl bit2 -> tgt bit1
    si ^= ((si >> 3) & 1) << 2;           // C(q0,q1): ctl bit3 -> tgt bit2
    float pr = 1.f, pi = 0.f;
    for (int q = 3; q >= 0; --q) {
      const float wr = ur[(si >> q) & 1][(j >> q) & 1];
      const float wi = ui[(si >> q) & 1][(j >> q) & 1];
      const float nr = pr * wr - pi * wi;
      const float ni = pr * wi + pi * wr;
      pr = nr; pi = ni;
    }
    g[t] = pr; g[256 + t] = pi; g[512 + t] = -pi;
  }
  // 4x4 group (wires 16,17): G2 = C(w16,w17) (U (x) U)
  if (t < 16) {
    const int i2 = t >> 2, j2 = t & 3;
    const int s2 = i2 ^ ((i2 >> 1) & 1);  // ctl bit1 -> tgt bit0
    float pr = 1.f, pi = 0.f;
    for (int q = 1; q >= 0; --q) {
      const float wr = ur[(s2 >> q) & 1][(j2 >> q) & 1];
      const float wi = ui[(s2 >> q) & 1][(j2 >> q) & 1];
      const float nr = pr * wr - pi * wi;
      const float ni = pr * wi + pi * wr;
      pr = nr; pi = ni;
    }
    g[768 + t] = pr; g[784 + t] = pi;
  }
}

// ---------------------------------------------------------------------------
// 4-wire group pass as complex GEMM via V_WMMA_F32_16X16X4_F32.
// Group nibble occupies flat bits [p, p+4). Columns enumerate
// (batch, upper, lower) = 2^20; one wave computes a 16(nib) x 16(col) tile.
// A layout (16x4 f32): lane m=L%16 holds row m; v0/v1 = K {0,1} (L<16) or
// {2,3} (L>=16). B (4x16): K rows split the same way, col = L%16.
// C/D (16x16 f32, 8 VGPRs): VGPR r = row r (L<16) / r+8 (L>=16).
// ---------------------------------------------------------------------------
__global__ __launch_bounds__(256) void qsim_group4(
    const float* __restrict__ sre, const float* __restrict__ sim,
    float* __restrict__ dre, float* __restrict__ dim_,
    const float* __restrict__ gates, int p, int applyPerm) {
  __shared__ float sg[768];
  const int tid = threadIdx.x;
  sg[tid]       = gates[tid];        // G4 re
  sg[256 + tid] = gates[256 + tid];  // G4 im
  sg[512 + tid] = gates[512 + tid];  // -G4 im
  __syncthreads();

  const int lane = tid & 31;
  const int wave = tid >> 5;
  const unsigned tile = blockIdx.x * 8u + (unsigned)wave;
  const unsigned col  = (tile << 4) + (unsigned)(lane & 15);
  const int kb = (lane >> 4) << 1;  // 0 for lanes 0-15, 2 for lanes 16-31
  const int m  = lane & 15;

  const unsigned pmask = (1u << p) - 1u;
  const unsigned low = col & pmask;
  const unsigned up  = (col >> p) & ((1u << (14 - p)) - 1u);
  const unsigned b   = col >> 14;
  const unsigned Fbase = (b << DIMBITS) | (up << (p + 4)) | low;

  // Each lane gathers its column's 8 K-slots (re+im). Optional CNOT-ring
  // permutation from the previous layer is folded into these addresses.
  float lre[8], lim[8];
#pragma unroll
  for (int j = 0; j < 8; ++j) {
    const int k = ((j >> 1) << 2) + kb + (j & 1);
    unsigned F = Fbase + ((unsigned)k << p);
    if (applyPerm) F = permIdx(F & 0x3FFFFu) | (F & ~0x3FFFFu);
    lre[j] = sre[F];
    lim[j] = sim[F];
  }

  v8f cre = {0.f, 0.f, 0.f, 0.f, 0.f, 0.f, 0.f, 0.f};
  v8f cim = {0.f, 0.f, 0.f, 0.f, 0.f, 0.f, 0.f, 0.f};
#pragma unroll
  for (int t = 0; t < 4; ++t) {
    const int k0 = t << 2;
    const int ga = m * 16 + k0 + kb;
    v2f ar  = {sg[ga],       sg[ga + 1]};        // G_re slice
    v2f ai  = {sg[256 + ga], sg[256 + ga + 1]};  // G_im slice
    v2f ain = {sg[512 + ga], sg[512 + ga + 1]};  // -G_im slice
    v2f br  = {lre[2 * t], lre[2 * t + 1]};
    v2f bi  = {lim[2 * t], lim[2 * t + 1]};
    // D_re = G_re*Psi_re - G_im*Psi_im ; D_im = G_re*Psi_im + G_im*Psi_re
    cre = __builtin_amdgcn_wmma_f32_16x16x4_f32(false, ar,  false, br, (short)0, cre, false, false);
    cre = __builtin_amdgcn_wmma_f32_16x16x4_f32(false, ain, false, bi, (short)0, cre, false, false);
    cim = __builtin_amdgcn_wmma_f32_16x16x4_f32(false, ar,  false, bi, (short)0, cim, false, false);
    cim = __builtin_amdgcn_wmma_f32_16x16x4_f32(false, ai,  false, br, (short)0, cim, false, false);
  }

  const unsigned rowOff = (unsigned)((lane >> 4) << 3);
#pragma unroll
  for (int r = 0; r < 8; ++r) {
    const unsigned F = Fbase + ((r + rowOff) << p);
    dre[F]  = cre[r];
    dim_[F] = cim[r];
  }
}

// ---------------------------------------------------------------------------
// 2-wire group pass (wires 16,17 = flat bits 1..0): 4x4 complex matvec on
// 4 consecutive amplitudes. Pure VALU; 1 of 5 passes per layer.
// ---------------------------------------------------------------------------
__global__ __launch_bounds__(256) void qsim_group2(
    const float* __restrict__ sre, const float* __restrict__ sim,
    float* __restrict__ dre, float* __restrict__ dim_,
    const float* __restrict__ gates) {
  __shared__ float g2r[16], g2i[16];
  const int tid = threadIdx.x;
  if (tid < 16) { g2r[tid] = gates[768 + tid]; g2i[tid] = gates[784 + tid]; }
  __syncthreads();

  const unsigned q = blockIdx.x * 256u + (unsigned)tid;  // quad id < 2^22
  const unsigned F = q << 2;
  const float4 r4 = *(const float4*)(sre + F);
  const float4 i4 = *(const float4*)(sim + F);
  const float rr[4] = {r4.x, r4.y, r4.z, r4.w};
  const float ii[4] = {i4.x, i4.y, i4.z, i4.w};
  float orr[4], oii[4];
#pragma unroll
  for (int i = 0; i < 4; ++i) {
    float ar = 0.f, ai = 0.f;
#pragma unroll
    for (int j = 0; j < 4; ++j) {
      ar = fmaf(g2r[4 * i + j], rr[j], ar);
      ar = fmaf(-g2i[4 * i + j], ii[j], ar);
      ai = fmaf(g2r[4 * i + j], ii[j], ai);
      ai = fmaf(g2i[4 * i + j], rr[j], ai);
    }
    orr[i] = ar; oii[i] = ai;
  }
  *(float4*)(dre + F)  = make_float4(orr[0], orr[1], orr[2], orr[3]);
  *(float4*)(dim_ + F) = make_float4(oii[0], oii[1], oii[2], oii[3]);
}

// ---------------------------------------------------------------------------
// Per-batch <Z_w> marginals + linear head. One block per batch element.
// Final layer's cross-group CNOT permutation is folded into the loads.
// ---------------------------------------------------------------------------
__global__ __launch_bounds__(256) void qsim_reduce(
    const float* __restrict__ sre, const float* __restrict__ sim,
    const float* __restrict__ hw, const float* __restrict__ hb,
    float* __restrict__ out) {
  __shared__ float red[256];
  __shared__ float zf[NW];
  const int tid = threadIdx.x;
  const unsigned b = blockIdx.x;
  const unsigned base = b << DIMBITS;

  float acc[NW];
#pragma unroll
  for (int w = 0; w < NW; ++w) acc[w] = 0.f;

  for (unsigned i = tid; i < (1u << DIMBITS); i += 256u) {
    const unsigned F = base + permIdx(i);
    const float re = sre[F], im = sim[F];
    const float pr = re * re + im * im;
#pragma unroll
    for (int w = 0; w < NW; ++w) {
      const float sgn = ((i >> (17 - w)) & 1u) ? -1.f : 1.f;
      acc[w] = fmaf(sgn, pr, acc[w]);
    }
  }

  for (int w = 0; w < NW; ++w) {
    red[tid] = acc[w];
    __syncthreads();
    for (int s = 128; s > 0; s >>= 1) {
      if (tid < s) red[tid] += red[tid + s];
      __syncthreads();
    }
    if (tid == 0) zf[w] = red[0];
    __syncthreads();
  }
  if (tid == 0) {
    float o = hb[0];
#pragma unroll
    for (int w = 0; w < NW; ++w) o = fmaf(zf[w], hw[w], o);
    out[b] = o;
  }
}

// ---------------------------------------------------------------------------
extern "C" void kernel_launch(void* const* d_in, const int* in_sizes, int n_in,
                              void* d_out, int out_size, void* d_ws, size_t ws_size,
                              hipStream_t stream) {
  (void)in_sizes; (void)n_in; (void)out_size; (void)ws_size;
  const float* in_re = (const float*)d_in[0];
  const float* in_im = (const float*)d_in[1];
  const float* hw    = (const float*)d_in[2];
  const float* hb    = (const float*)d_in[3];
  float* out = (float*)d_out;

  float* w = (float*)d_ws;
  const size_t SZ = (size_t)BATCH << DIMBITS;  // floats per plane
  float* gates = w;
  float* A_re = w + 1024;
  float* A_im = A_re + SZ;
  float* B_re = A_im + SZ;
  float* B_im = B_re + SZ;

  qsim_build<<<1, 256, 0, stream>>>(gates);

  const float* sr = in_re;
  const float* si = in_im;
  int cur = 0;
  for (int layer = 0; layer < 3; ++layer) {
    for (int g4 = 0; g4 < 4; ++g4) {
      const int p = 14 - 4 * g4;                       // nibble LSB position
      const int perm = (layer > 0 && g4 == 0) ? 1 : 0;  // fold prev ring CNOTs
      float* dr  = cur ? B_re : A_re;
      float* di_ = cur ? B_im : A_im;
      // 2^20 cols / 16 per wave = 65536 waves; 8 waves per 256-thread block.
      qsim_group4<<<8192, 256, 0, stream>>>(sr, si, dr, di_, gates, p, perm);
      sr = dr; si = di_; cur ^= 1;
    }
    float* dr  = cur ? B_re : A_re;
    float* di_ = cur ? B_im : A_im;
    qsim_group2<<<16384, 256, 0, stream>>>(sr, si, dr, di_, gates);
    sr = dr; si = di_; cur ^= 1;
  }
  qsim_reduce<<<BATCH, 256, 0, stream>>>(sr, si, hw, hb, out);
}